// IndGRU_45535243272432
// MI455X (gfx1250) — compile-verified
//
#include <hip/hip_runtime.h>

// ---------------------------------------------------------------------------
// IndGRU for MI455X (gfx1250, wave32):
//   Phase 1: gx[t,b,g] = sum_i x[t,b,i] * W_ih[g,i]   (M=16384, K=512, N=1536)
//            -> fp32 WMMA (V_WMMA_F32_16X16X4_F32), exact fp32 like reference.
//            A in LDS [row][k] (stride 36: bank-conflict-free, 16B aligned),
//            B in LDS K-paired layout [k/2][n][2] (stride 160 dwords) so each
//            B fragment is a single conflict-free ds_load_b64 (no repacking).
//   Phase 2: diagonal-recurrence scan, 1 thread per (b,h) channel, T=512 steps,
//            gx (96MB) stays resident in the 192MB L2, loads software-pipelined.
// ---------------------------------------------------------------------------

typedef float v2f __attribute__((ext_vector_type(2)));
typedef float v8f __attribute__((ext_vector_type(8)));

#define T_DIM 512
#define B_DIM 32
#define I_DIM 512
#define H_DIM 512
#define M_DIM (T_DIM * B_DIM)   // 16384 rows (t*B + b)
#define N_DIM (3 * H_DIM)       // 1536 cols (gate-major: r | z | n)
#define K_DIM I_DIM             // 512

#define BM 128
#define BN 64
#define BK 32
#define LDA 36    // A LDS row stride (floats): 36*r mod 64 distinct for 16 rows
#define LDBP 160  // B LDS k-pair row stride (floats): +160 -> bank shift of 32

__global__ __launch_bounds__(256)
void indgru_gemm_wmma(const float* __restrict__ x, const float* __restrict__ W,
                      float* __restrict__ gx) {
    __shared__ float sA[BM * LDA];            // [row][k]
    __shared__ float sB[(BK / 2) * LDBP];     // [k/2][n][2]

    const int tid  = threadIdx.x;
    const int wid  = tid >> 5;       // wave id 0..7 (wave32)
    const int lane = tid & 31;
    const int lo   = lane & 15;
    const int hi   = lane >> 4;      // K half-select per ISA f32 A/B layout

    const int m_blk = blockIdx.x * BM;
    const int n_blk = blockIdx.y * BN;

    const int wm = (wid & 3) * 32;   // wave M offset (4 waves in M)
    const int wn = (wid >> 2) * 32;  // wave N offset (2 waves in N)

    v8f c00 = {}, c01 = {}, c10 = {}, c11 = {};

    // staging assignments (coalesced float4 global loads)
    const int rowA = tid >> 1;            // 0..127
    const int colA = (tid & 1) * 16;      // 0 or 16
    const int nB   = tid >> 2;            // 0..63  (row of W = output col)
    const int kB   = (tid & 3) * 8;       // 0,8,16,24

    for (int k0 = 0; k0 < K_DIM; k0 += BK) {
        // ---- stage A tile: BM x BK, row-major, 16 floats/thread ----
        const float4* gA = (const float4*)(x + (size_t)(m_blk + rowA) * K_DIM + k0 + colA);
        float4* lA = (float4*)(sA + rowA * LDA + colA);
        lA[0] = gA[0]; lA[1] = gA[1]; lA[2] = gA[2]; lA[3] = gA[3];

        // ---- stage B tile: read W rows (n-major), write K-paired layout ----
        // element B[k][n] lives at sB[(k>>1)*LDBP + n*2 + (k&1)]
        const float4* gB = (const float4*)(W + (size_t)(n_blk + nB) * K_DIM + k0 + kB);
        float4 w0 = gB[0], w1 = gB[1];
        float* pB = sB + (kB >> 1) * LDBP + nB * 2;
        *(v2f*)(pB + 0 * LDBP) = (v2f){w0.x, w0.y};
        *(v2f*)(pB + 1 * LDBP) = (v2f){w0.z, w0.w};
        *(v2f*)(pB + 2 * LDBP) = (v2f){w1.x, w1.y};
        *(v2f*)(pB + 3 * LDBP) = (v2f){w1.z, w1.w};
        __syncthreads();

        #pragma unroll
        for (int kk = 0; kk < BK; kk += 4) {
            const int kp = (kk >> 1) + hi;   // k-pair row: hi lanes take K=2,3
            // A fragments (two 16-row subtiles): {A[m][ka], A[m][ka+1]}
            v2f a0 = *(const v2f*)(sA + (wm +      lo) * LDA + 2 * kp);
            v2f a1 = *(const v2f*)(sA + (wm + 16 + lo) * LDA + 2 * kp);
            // B fragments (two 16-col subtiles): single ds_load_b64 each
            v2f b0 = *(const v2f*)(sB + kp * LDBP + (wn +      lo) * 2);
            v2f b1 = *(const v2f*)(sB + kp * LDBP + (wn + 16 + lo) * 2);

            c00 = __builtin_amdgcn_wmma_f32_16x16x4_f32(false, a0, false, b0, (short)0, c00, false, false);
            c01 = __builtin_amdgcn_wmma_f32_16x16x4_f32(false, a0, false, b1, (short)0, c01, false, false);
            c10 = __builtin_amdgcn_wmma_f32_16x16x4_f32(false, a1, false, b0, (short)0, c10, false, false);
            c11 = __builtin_amdgcn_wmma_f32_16x16x4_f32(false, a1, false, b1, (short)0, c11, false, false);
        }
        __syncthreads();
    }

    // ---- epilogue: C layout (ISA 7.12.2): VGPR j -> M=j (lanes 0-15) / M=8+j ----
    const int cn0 = n_blk + wn + lo;
    #pragma unroll
    for (int j = 0; j < 8; ++j) {
        const int rm0 = m_blk + wm + hi * 8 + j;
        gx[(size_t)rm0 * N_DIM + cn0]             = c00[j];
        gx[(size_t)rm0 * N_DIM + cn0 + 16]        = c01[j];
        gx[(size_t)(rm0 + 16) * N_DIM + cn0]      = c10[j];
        gx[(size_t)(rm0 + 16) * N_DIM + cn0 + 16] = c11[j];
    }
}

// ---------------------------------------------------------------------------
// Phase 2: sequential scan. h_t = (1-z)*n + z*h, all elementwise per channel.
// ---------------------------------------------------------------------------
__device__ __forceinline__ float fast_sigmoid(float v) {
    return __builtin_amdgcn_rcpf(1.0f + __expf(-v));   // v_exp_f32 + v_rcp_f32
}
__device__ __forceinline__ float fast_tanh(float v) {
    float e = __expf(2.0f * v);                        // saturates correctly at +/-inf
    return 1.0f - 2.0f * __builtin_amdgcn_rcpf(e + 1.0f);
}

__global__ __launch_bounds__(256)
void indgru_scan(const float* __restrict__ gx, const float* __restrict__ h0,
                 const float* __restrict__ w_hh, float* __restrict__ out,
                 float* __restrict__ hlast) {
    const int idx = blockIdx.x * 256 + threadIdx.x;    // 0 .. B*H-1
    const int b = idx >> 9;
    const int h = idx & (H_DIM - 1);

    const float wr = w_hh[h];
    const float wz = w_hh[H_DIM + h];
    const float wn = w_hh[2 * H_DIM + h];
    float hc = h0[idx];

    const float* g = gx + (size_t)b * N_DIM + h;       // step-t gate pointer
    float*       o = out + (size_t)b * H_DIM + h;
    const size_t gstep = (size_t)B_DIM * N_DIM;
    const size_t ostep = (size_t)B_DIM * H_DIM;

    // software pipeline: issue t+1 loads before computing step t (L2-resident gx)
    float gr = g[0], gz = g[H_DIM], gn = g[2 * H_DIM];
    for (int t = 0; t < T_DIM - 1; ++t) {
        const float* gnx = g + gstep;
        const float nr = gnx[0];
        const float nz = gnx[H_DIM];
        const float nn = gnx[2 * H_DIM];

        const float r = fast_sigmoid(gr + wr * hc);
        const float z = fast_sigmoid(gz + wz * hc);
        const float n = fast_tanh(gn + r * (wn * hc));
        hc = (1.0f - z) * n + z * hc;
        *o = hc;

        gr = nr; gz = nz; gn = nn;
        g = gnx; o += ostep;
    }
    // final step (no prefetch)
    {
        const float r = fast_sigmoid(gr + wr * hc);
        const float z = fast_sigmoid(gz + wz * hc);
        const float n = fast_tanh(gn + r * (wn * hc));
        hc = (1.0f - z) * n + z * hc;
        *o = hc;
    }
    hlast[idx] = hc;
}

// ---------------------------------------------------------------------------
extern "C" void kernel_launch(void* const* d_in, const int* in_sizes, int n_in,
                              void* d_out, int out_size, void* d_ws, size_t ws_size,
                              hipStream_t stream) {
    const float* x    = (const float*)d_in[0];   // (T,B,I)
    const float* h0   = (const float*)d_in[1];   // (B,H)
    const float* W_ih = (const float*)d_in[2];   // (3H,I)
    const float* w_hh = (const float*)d_in[3];   // (3,H)

    float* out   = (float*)d_out;                          // (T,B,H)
    float* hlast = out + (size_t)T_DIM * B_DIM * H_DIM;    // (1,B,H) appended
    float* gx    = (float*)d_ws;                           // (T,B,3H) = 96 MB scratch

    dim3 ggrid(M_DIM / BM, N_DIM / BN);                    // 128 x 24
    indgru_gemm_wmma<<<ggrid, 256, 0, stream>>>(x, W_ih, gx);

    indgru_scan<<<(B_DIM * H_DIM) / 256, 256, 0, stream>>>(gx, h0, w_hh, out, hlast);
}